// HierarchicalSoftmax_87737591922854
// MI455X (gfx1250) — compile-verified
//
#include <hip/hip_runtime.h>
#include <math.h>

typedef __attribute__((ext_vector_type(2))) float v2f;
typedef __attribute__((ext_vector_type(8))) float v8f;

#define NHID  512
#define H0    256
#define H1    256
#define BATCH 2048

#define XS_PITCH 516   // 512 + 4: LDS bank-conflict padding for A reads
#define LG_PITCH 273   // 256 + 17: conflict-free logit store + softmax read

__device__ __forceinline__ float red16_max(float v) {
  #pragma unroll
  for (int off = 8; off; off >>= 1) v = fmaxf(v, __shfl_xor(v, off, 16));
  return v;
}
__device__ __forceinline__ float red16_sum(float v) {
  #pragma unroll
  for (int off = 8; off; off >>= 1) v += __shfl_xor(v, off, 16);
  return v;
}

// ---------------- grouping (counting sort by parent) ----------------
__global__ void hs_zero_kernel(int* __restrict__ counts, int* __restrict__ cursors) {
  int i = threadIdx.x;
  counts[i] = 0;
  cursors[i] = 0;
}
__global__ void hs_count_kernel(const int* __restrict__ parent, int* __restrict__ counts) {
  int b = blockIdx.x * blockDim.x + threadIdx.x;
  atomicAdd(&counts[parent[b]], 1);
}
__global__ void hs_scan_kernel(const int* __restrict__ counts, int* __restrict__ offsets) {
  if (threadIdx.x == 0) {
    int acc = 0;
    for (int i = 0; i < H0; ++i) { offsets[i] = acc; acc += counts[i]; }
  }
}
__global__ void hs_scatter_kernel(const int* __restrict__ parent,
                                  const int* __restrict__ offsets,
                                  int* __restrict__ cursors,
                                  int* __restrict__ perm) {
  int b = blockIdx.x * blockDim.x + threadIdx.x;
  int p = parent[b];
  int pos = offsets[p] + atomicAdd(&cursors[p], 1);
  perm[pos] = b;
}

// ---------------- top GEMM + softmax target prob ----------------
__global__ __launch_bounds__(256) void hs_top_kernel(
    const float* __restrict__ X, const float* __restrict__ Wt,
    const float* __restrict__ bt, const int* __restrict__ parent,
    float* __restrict__ top_target) {
  __shared__ float xs[16][XS_PITCH];
  __shared__ float lg[16][LG_PITCH];
  const int m0  = blockIdx.x * 16;
  const int tid = threadIdx.x;

  // stage 16 rows of X into LDS (coalesced float4)
  for (int linear = tid * 4; linear < 16 * NHID; linear += 256 * 4) {
    int r = linear >> 9;
    int c = linear & (NHID - 1);
    *(float4*)&xs[r][c] = *(const float4*)&X[(size_t)(m0 + r) * NHID + c];
  }
  __syncthreads();

  const int lane = tid & 31;
  const int wv   = tid >> 5;
  const int col  = lane & 15;            // A row (M) and B column (N) sub-index
  const int koff = (lane >> 4) * 2;      // K slot: lanes 0-15 -> K{0,1}, 16-31 -> K{2,3}
  const int n0   = wv * 32;
  const int n1   = n0 + 16;

  v8f acc0 = {}, acc1 = {};
  #pragma unroll 4
  for (int k0 = 0; k0 < NHID; k0 += 4) {
    v2f a = *(const v2f*)&xs[col][k0 + koff];
    const float* w0 = &Wt[(size_t)(k0 + koff) * H0];
    v2f b0 = { w0[n0 + col], w0[H0 + n0 + col] };
    v2f b1 = { w0[n1 + col], w0[H0 + n1 + col] };
    acc0 = __builtin_amdgcn_wmma_f32_16x16x4_f32(false, a, false, b0, (short)0, acc0, false, false);
    acc1 = __builtin_amdgcn_wmma_f32_16x16x4_f32(false, a, false, b1, (short)0, acc1, false, false);
  }

  // bias + write logits to LDS (VGPR j -> M = j + 8*(lane>=16))
  const int rbase = (lane >> 4) * 8;
  const float bias0 = bt[n0 + col];
  const float bias1 = bt[n1 + col];
  #pragma unroll
  for (int j = 0; j < 8; ++j) {
    lg[rbase + j][n0 + col] = acc0[j] + bias0;
    lg[rbase + j][n1 + col] = acc1[j] + bias1;
  }
  __syncthreads();

  // per-row softmax: wave wv owns rows 2*wv (lanes 0-15) and 2*wv+1 (lanes 16-31)
  const int row = wv * 2 + (lane >> 4);
  const int j16 = lane & 15;
  float m = -INFINITY;
  for (int c = j16; c < H0; c += 16) m = fmaxf(m, lg[row][c]);
  m = red16_max(m);
  float s = 0.f;
  for (int c = j16; c < H0; c += 16) s += __expf(lg[row][c] - m);
  s = red16_sum(s);
  if (j16 == 0) {
    const int b = m0 + row;
    const int p = parent[b];
    top_target[b] = __expf(lg[row][p] - m) / s;
  }
}

// ---------------- bottom: one block per parent (grouped GEMM) ----------------
__global__ __launch_bounds__(256) void hs_bottom_kernel(
    const float* __restrict__ X, const float* __restrict__ Wb,
    const float* __restrict__ bb, const int* __restrict__ labels,
    const int* __restrict__ counts, const int* __restrict__ offsets,
    const int* __restrict__ perm, const float* __restrict__ top_target,
    float* __restrict__ out) {
  __shared__ float xs[16][XS_PITCH];
  __shared__ float lg[16][LG_PITCH];
  __shared__ int   rowsamp[16];

  const int p     = blockIdx.x;
  const int nrows = counts[p];
  if (nrows == 0) return;                // block-uniform
  const int base  = offsets[p];
  const int tid   = threadIdx.x;
  const int lane  = tid & 31;
  const int wv    = tid >> 5;
  const int col   = lane & 15;
  const int koff  = (lane >> 4) * 2;
  const int n0    = wv * 32;
  const int n1    = n0 + 16;

  const float* Wp   = Wb + (size_t)p * NHID * H1;
  const float bias0 = bb[(size_t)p * H1 + n0 + col];
  const float bias1 = bb[(size_t)p * H1 + n1 + col];

  for (int r0 = 0; r0 < nrows; r0 += 16) {
    if (tid < 16) {
      int r = r0 + tid;
      rowsamp[tid] = (r < nrows) ? perm[base + r] : -1;
    }
    __syncthreads();

    // stage grouped rows (zero-pad invalid tile rows)
    for (int linear = tid * 4; linear < 16 * NHID; linear += 256 * 4) {
      int r = linear >> 9;
      int c = linear & (NHID - 1);
      int b = rowsamp[r];
      float4 v = make_float4(0.f, 0.f, 0.f, 0.f);
      if (b >= 0) v = *(const float4*)&X[(size_t)b * NHID + c];
      *(float4*)&xs[r][c] = v;
    }
    __syncthreads();

    v8f acc0 = {}, acc1 = {};
    #pragma unroll 4
    for (int k0 = 0; k0 < NHID; k0 += 4) {
      v2f a = *(const v2f*)&xs[col][k0 + koff];
      const float* w0 = &Wp[(size_t)(k0 + koff) * H1];
      v2f b0 = { w0[n0 + col], w0[H1 + n0 + col] };
      v2f b1 = { w0[n1 + col], w0[H1 + n1 + col] };
      acc0 = __builtin_amdgcn_wmma_f32_16x16x4_f32(false, a, false, b0, (short)0, acc0, false, false);
      acc1 = __builtin_amdgcn_wmma_f32_16x16x4_f32(false, a, false, b1, (short)0, acc1, false, false);
    }

    const int rbase = (lane >> 4) * 8;
    #pragma unroll
    for (int j = 0; j < 8; ++j) {
      lg[rbase + j][n0 + col] = acc0[j] + bias0;
      lg[rbase + j][n1 + col] = acc1[j] + bias1;
    }
    __syncthreads();

    const int row = wv * 2 + (lane >> 4);
    const int j16 = lane & 15;
    float m = -INFINITY;
    for (int c = j16; c < H1; c += 16) m = fmaxf(m, lg[row][c]);
    m = red16_max(m);
    float s = 0.f;
    for (int c = j16; c < H1; c += 16) s += __expf(lg[row][c] - m);
    s = red16_sum(s);
    if (j16 == 0) {
      int b = rowsamp[row];
      if (b >= 0) {
        int lab = labels[b];
        out[b] = top_target[b] * (__expf(lg[row][lab] - m) / s);
      }
    }
    __syncthreads();   // protect xs/lg/rowsamp before next row-tile
  }
}

extern "C" void kernel_launch(void* const* d_in, const int* in_sizes, int n_in,
                              void* d_out, int out_size, void* d_ws, size_t ws_size,
                              hipStream_t stream) {
  const float* X      = (const float*)d_in[0];
  const int*   labels = (const int*)d_in[1];
  const int*   parent = (const int*)d_in[2];
  const float* Wt     = (const float*)d_in[3];
  const float* bt     = (const float*)d_in[4];
  const float* Wb     = (const float*)d_in[5];
  const float* bb     = (const float*)d_in[6];
  float* out = (float*)d_out;

  char* ws = (char*)d_ws;
  float* top_target = (float*)ws;                        // 2048 floats
  int*   counts     = (int*)(ws + BATCH * 4);            // 256 ints
  int*   offsets    = (int*)(ws + BATCH * 4 + 256 * 4);  // 256 ints
  int*   cursors    = (int*)(ws + BATCH * 4 + 512 * 4);  // 256 ints
  int*   perm       = (int*)(ws + BATCH * 4 + 768 * 4);  // 2048 ints

  hs_zero_kernel   <<<1, H0, 0, stream>>>(counts, cursors);
  hs_count_kernel  <<<BATCH / 256, 256, 0, stream>>>(parent, counts);
  hs_scan_kernel   <<<1, 32, 0, stream>>>(counts, offsets);
  hs_scatter_kernel<<<BATCH / 256, 256, 0, stream>>>(parent, offsets, cursors, perm);
  hs_top_kernel    <<<BATCH / 16, 256, 0, stream>>>(X, Wt, bt, parent, top_target);
  hs_bottom_kernel <<<H0, 256, 0, stream>>>(X, Wb, bb, labels, counts, offsets, perm,
                                            top_target, out);
}